// Cont_Loss_2705829397112
// MI455X (gfx1250) — compile-verified
//
#include <hip/hip_runtime.h>

typedef __attribute__((ext_vector_type(2))) float v2f;
typedef __attribute__((ext_vector_type(8))) float v8f;

#define SLEN   4096
#define FEAT   256
#define NBATCH 4
#define KTOT   (NBATCH * FEAT)        // 1024
#define CHUNK  ((size_t)SLEN * FEAT)  // per-batch stride in floats

// ws float layout: s[0..4095] | colsum[4096..5119] | adjd[5120..9215]
#define WS_S    0
#define WS_COL  SLEN
#define WS_ADJ  (SLEN + KTOT)

// ---------- colsum[j] = sum_t x[n,t,f] for j = n*256+f  (one thread per column)
__global__ void k_colsum(const float* __restrict__ x, float* __restrict__ ws) {
    int j = blockIdx.x * blockDim.x + threadIdx.x;   // 0..1023
    int n = j >> 8, f = j & 255;
    const float* p = x + (size_t)n * CHUNK + f;
    float acc = 0.f;
    for (int t = 0; t < SLEN; ++t) acc += p[(size_t)t * FEAT];
    ws[WS_COL + j] = acc;
}

// ---------- s[t] = ||v_t||^2  (one wave32 per t, coalesced, shfl reduce)
__global__ void k_rownorm(const float* __restrict__ x, float* __restrict__ ws) {
    int lane = threadIdx.x & 31;
    int wv   = threadIdx.x >> 5;
    int t    = blockIdx.x * 8 + wv;                  // 512 blocks * 8 waves = 4096
    float acc = 0.f;
    for (int j = lane; j < KTOT; j += 32) {
        float v = x[(size_t)(j >> 8) * CHUNK + (size_t)t * FEAT + (j & 255)];
        acc += v * v;
    }
    for (int off = 16; off; off >>= 1) acc += __shfl_xor(acc, off, 32);
    if (lane == 0) ws[WS_S + t] = acc;
}

// ---------- adjd[a] = <v_a, v_{a-1}> via V_WMMA_F32_16X16X4_F32 (exact fp32)
// One wave per 16-row group k: block G[16k..16k+15, 16k-1..16k+14], diagonal = adjacent dots.
__global__ void k_adj_wmma(const float* __restrict__ x, float* __restrict__ ws) {
    int lane = threadIdx.x & 31;
    int wv   = threadIdx.x >> 5;
    int k    = blockIdx.x * 8 + wv;                  // 32 blocks * 8 waves = 256 groups
    int m    = lane & 15;
    int half = lane >> 4;                            // selects K pair {0,1} vs {2,3}
    int rowA = k * 16 + m;                           // A: lane = M
    int rowB = k * 16 - 1 + m;                       // B: lane = N (shifted by one row)
    if (rowB < 0) rowB = 0;                          // k=0 lane m=0: dummy, result discarded

    v8f c = {};
    for (int n = 0; n < NBATCH; ++n) {
        const float* baseA = x + (size_t)n * CHUNK + (size_t)rowA * FEAT + 2 * half;
        const float* baseB = x + (size_t)n * CHUNK + (size_t)rowB * FEAT + 2 * half;
        #pragma unroll 4
        for (int kk = 0; kk < FEAT; kk += 4) {       // 64 WMMAs per chunk, 256 total
            v2f a = *(const v2f*)(baseA + kk);       // A 16x4 f32: vgpr0=K0/K2, vgpr1=K1/K3
            v2f b = *(const v2f*)(baseB + kk);       // B 4x16 f32: mirrored layout
            c = __builtin_amdgcn_wmma_f32_16x16x4_f32(
                    /*neg_a=*/false, a, /*neg_b=*/false, b,
                    /*c_mod=*/(short)0, c, /*reuse_a=*/false, /*reuse_b=*/false);
        }
    }

    // Diagonal (i,i): i<8 -> lane i, vgpr i ; i>=8 -> lane 16+i, vgpr i-8
    int idx = -1, i = 0;
    if (lane < 8)        { idx = lane;      i = lane;      }
    else if (lane >= 24) { idx = lane - 24; i = lane - 16; }
    if (idx >= 0) {
        union { v8f v; float f[8]; } u; u.v = c;
        int a_glob = k * 16 + i;                     // pair (a_glob, a_glob-1)
        if (a_glob >= 1) ws[WS_ADJ + a_glob] = u.f[idx];
    }
}

// ---------- final: loss = (S*sum_s - ||colsum||^2 - sum_adj D + sum_adj hinge) / (S*(S-1)*1000)
__global__ void k_final(const float* __restrict__ ws, float* __restrict__ out) {
    __shared__ double sred[256], nred[256], dred[256], hred[256];
    int tid = threadIdx.x;
    double ps = 0, pn = 0, pd = 0, ph = 0;
    for (int t = tid; t < SLEN; t += 256) ps += (double)ws[WS_S + t];
    for (int j = tid; j < KTOT; j += 256) { double v = ws[WS_COL + j]; pn += v * v; }
    for (int a = 1 + tid; a < SLEN; a += 256) {
        double Dadj = (double)ws[WS_S + a] + (double)ws[WS_S + a - 1]
                      - 2.0 * (double)ws[WS_ADJ + a];
        pd += Dadj;
        double h = 60000.0 - Dadj;
        ph += (h > 0.0) ? h : 0.0;
    }
    sred[tid] = ps; nred[tid] = pn; dred[tid] = pd; hred[tid] = ph;
    __syncthreads();
    for (int off = 128; off; off >>= 1) {
        if (tid < off) {
            sred[tid] += sred[tid + off]; nred[tid] += nred[tid + off];
            dred[tid] += dred[tid + off]; hred[tid] += hred[tid + off];
        }
        __syncthreads();
    }
    if (tid == 0) {
        double sum_s = sred[0], norm2 = nred[0];
        double Dlower = (double)SLEN * sum_s - norm2;      // sum over all a>b of D
        double loss   = Dlower - dred[0] + hred[0];        // swap adjacent D -> hinge
        out[0] = (float)(loss / ((double)SLEN * (SLEN - 1) * 1000.0));
    }
}

extern "C" void kernel_launch(void* const* d_in, const int* in_sizes, int n_in,
                              void* d_out, int out_size, void* d_ws, size_t ws_size,
                              hipStream_t stream) {
    const float* x = (const float*)d_in[0];
    float* ws = (float*)d_ws;                        // needs (4096+1024+4096)*4 = 36,864 B
    k_colsum <<<KTOT / 256, 256, 0, stream>>>(x, ws);
    k_rownorm<<<SLEN / 8,   256, 0, stream>>>(x, ws);
    k_adj_wmma<<<256 / 8,   256, 0, stream>>>(x, ws);
    k_final  <<<1,          256, 0, stream>>>(ws, (float*)d_out);
}